// CGMSA_84344567758975
// MI455X (gfx1250) — compile-verified
//
#include <hip/hip_runtime.h>
#include <hip/hip_bf16.h>

// ---------------------------------------------------------------------------
// CGMSA for MI455X (gfx1250): fp32 WMMA (v_wmma_f32_16x16x4_f32) everywhere.
// Memory-bound workload (~56 GFLOP vs ~0.6 GB HBM traffic) -> keep fp32.
// ---------------------------------------------------------------------------

typedef __attribute__((ext_vector_type(2))) float v2f;
typedef __attribute__((ext_vector_type(8))) float v8f;

#define BATCH 8
#define CIN   192
#define C2    384
#define HH    120
#define WW    120
#define HWPIX 14400

static __device__ __forceinline__ v8f wmma_f32_k4(v2f a, v2f b, v8f c) {
    // D = A(16x4) * B(4x16) + C(16x16), all fp32, wave32
    return __builtin_amdgcn_wmma_f32_16x16x4_f32(
        false, a, false, b, (short)0, c, false, false);
}

// ---------------------------------------------------------------------------
// Fold batchnorm into conv_in weight/bias.
// ---------------------------------------------------------------------------
__global__ void fold_bn_kernel(const float* __restrict__ w_in,
                               const float* __restrict__ b_in,
                               const float* __restrict__ gamma,
                               const float* __restrict__ beta,
                               const float* __restrict__ mean,
                               const float* __restrict__ var,
                               float* __restrict__ wbn,
                               float* __restrict__ biasbn) {
    int oc = blockIdx.x * blockDim.x + threadIdx.x;
    if (oc >= C2) return;
    float sc = gamma[oc] * rsqrtf(var[oc] + 1e-5f);
    biasbn[oc] = (b_in[oc] - mean[oc]) * sc + beta[oc];
    const float* wr = w_in + (size_t)oc * CIN;
    float* wo = wbn + (size_t)oc * CIN;
#pragma unroll 4
    for (int c = 0; c < CIN; ++c) wo[c] = wr[c] * sc;
}

// ---------------------------------------------------------------------------
// 1x1 conv as per-batch GEMM: out[b][m][p] = sum_k W[m][k] * in[b][k][p] + bias[m]
// Channel rows are contiguous (stride HWPIX) -> B-operand lanes map to pixels.
// Grid: (HWPIX/64, M/64, BATCH). Block: 128 (4 waves). Each wave: 16(M)x64(N).
// ---------------------------------------------------------------------------
__global__ void gemm1x1_kernel(float* __restrict__ out,
                               const float* __restrict__ in,
                               const float* __restrict__ Wm,
                               const float* __restrict__ bias,
                               int M, int K) {
    const int lane  = threadIdx.x & 31;
    const int wave  = threadIdx.x >> 5;
    const int lm    = lane & 15;
    const int khalf = (lane < 16) ? 0 : 2;   // A/B lane K-offset (ISA 7.12.2)
    const int n0    = blockIdx.x * 64;
    const int m0    = blockIdx.y * 64 + wave * 16;
    const int bb    = blockIdx.z;

    const float* inB  = in  + (size_t)bb * K * HWPIX;
    float*       outB = out + (size_t)bb * M * HWPIX;

    v8f acc[4] = {};
    for (int k0 = 0; k0 < K; k0 += 4) {
        // A fragment: W rows m0..m0+15, K slice k0..k0+3
        const float* ap = Wm + (size_t)(m0 + lm) * K + k0 + khalf;
        v2f a; a.x = ap[0]; a.y = ap[1];
        // prefetch next K-rows of activations into cache
        if (k0 + 4 < K)
            __builtin_prefetch(inB + (size_t)(k0 + 4 + khalf) * HWPIX + n0 + lm, 0, 1);
#pragma unroll
        for (int nt = 0; nt < 4; ++nt) {
            const float* bp = inB + (size_t)(k0 + khalf) * HWPIX + n0 + nt * 16 + lm;
            v2f b; b.x = bp[0]; b.y = bp[HWPIX];
            acc[nt] = wmma_f32_k4(a, b, acc[nt]);
        }
    }
    const int rowhi = (lane >> 4) * 8;
#pragma unroll
    for (int r = 0; r < 8; ++r) {
        int m = m0 + rowhi + r;
        float bv = bias[m];
#pragma unroll
        for (int nt = 0; nt < 4; ++nt)
            outB[(size_t)m * HWPIX + n0 + nt * 16 + lm] = acc[nt][r] + bv;
    }
}

// ---------------------------------------------------------------------------
// Windowed attention for one scale. One workgroup per window, T = L/16 waves,
// wave w owns rows [16w, 16w+16) of S = qx * qy^T (L x L), softmax in the
// C-fragment layout via shfl_xor over 16-lane halves, then O = atn * vx.
// Roll(-s) folded into gather indices; merge+roll(+s) folded into scatter.
// ---------------------------------------------------------------------------
template <int WS>
__global__ void attn_kernel(const float* __restrict__ xp,
                            const float* __restrict__ yp,
                            float* __restrict__ atn_out,
                            float* __restrict__ ocat,
                            int chunkIdx) {
    constexpr int L  = WS * WS;
    constexpr int T  = L / 16;
    constexpr int SH = WS / 2;
    constexpr int NH = HH / WS;
    constexpr int NW = WW / WS;

    extern __shared__ float lds[];
    float* atn_lds = lds;            // L*L
    float* v_lds   = lds + L * L;    // L*64

    const int n   = blockIdx.x;
    const int bb  = n / (NH * NW);
    const int rem = n - bb * (NH * NW);
    const int wi  = rem / NW;
    const int wj  = rem - wi * NW;

    const int lane  = threadIdx.x & 31;
    const int wave  = threadIdx.x >> 5;       // strip id 0..T-1
    const int lm    = lane & 15;
    const int khalf = (lane < 16) ? 0 : 2;
    const int rowhi = (lane >> 4) * 8;

    const int cq = chunkIdx * 128;            // q channels in xp/yp
    const int cv = cq + 64;                   // v channels
    const float* xb = xp + (size_t)bb * C2 * HWPIX;
    const float* yb = yp + (size_t)bb * C2 * HWPIX;

    // window element -> rolled global pixel (roll by -SH folded in)
    auto pix = [&](int k) -> int {
        int dy = k / WS, dx = k - dy * WS;
        int h = wi * WS + dy + SH; if (h >= HH) h -= HH;
        int w = wj * WS + dx + SH; if (w >= WW) w -= WW;
        return h * WW + w;
    };

    // ---- phase 0: stage vx into LDS  (v_lds[k][c], row-major 64) ----
    for (int idx = threadIdx.x; idx < L * 64; idx += blockDim.x) {
        int k = idx >> 6, c = idx & 63;
        v_lds[idx] = xb[(size_t)(cv + c) * HWPIX + pix(k)];
    }

    // ---- phase 1: S = qx * qy^T for this wave's 16-row strip ----
    const int m0 = wave * 16;
    const int myPix = pix(m0 + lm);           // pixel of my A-row
    v2f afr[16];                               // qx strip, all K=64
#pragma unroll
    for (int kk = 0; kk < 16; ++kk) {
        const float* ap = xb + (size_t)(cq + kk * 4 + khalf) * HWPIX + myPix;
        afr[kk].x = ap[0]; afr[kk].y = ap[HWPIX];
    }
    v8f accS[T] = {};
    for (int t = 0; t < T; ++t) {
        int np = pix(t * 16 + lm);            // pixel of B-column
#pragma unroll
        for (int kk = 0; kk < 16; ++kk) {
            const float* bp = yb + (size_t)(cq + kk * 4 + khalf) * HWPIX + np;
            v2f b; b.x = bp[0]; b.y = bp[HWPIX];
            accS[t] = wmma_f32_k4(afr[kk], b, accS[t]);
        }
    }

    // ---- softmax over columns, rows live in C-layout ----
    float mx[8], sm[8];
#pragma unroll
    for (int r = 0; r < 8; ++r) {
        float m = accS[0][r];
        for (int t = 1; t < T; ++t) m = fmaxf(m, accS[t][r]);
        mx[r] = m;
    }
#pragma unroll
    for (int mask = 8; mask >= 1; mask >>= 1)
#pragma unroll
        for (int r = 0; r < 8; ++r)
            mx[r] = fmaxf(mx[r], __shfl_xor(mx[r], mask, 32));
#pragma unroll
    for (int r = 0; r < 8; ++r) sm[r] = 0.f;
    for (int t = 0; t < T; ++t)
#pragma unroll
        for (int r = 0; r < 8; ++r) {
            float e = __expf(accS[t][r] - mx[r]);
            accS[t][r] = e; sm[r] += e;
        }
#pragma unroll
    for (int mask = 8; mask >= 1; mask >>= 1)
#pragma unroll
        for (int r = 0; r < 8; ++r)
            sm[r] += __shfl_xor(sm[r], mask, 32);

    float* atn_g = atn_out + (size_t)n * L * L;
    for (int t = 0; t < T; ++t) {
        int col = t * 16 + lm;
#pragma unroll
        for (int r = 0; r < 8; ++r) {
            int row = m0 + rowhi + r;
            float v = accS[t][r] / sm[r];
            atn_lds[row * L + col] = v;
            atn_g[row * L + col]   = v;
        }
    }
    __syncthreads();

    // ---- phase 2: O = atn * vx  (M=16 strip, N=64, K=L) ----
    v8f accO[4] = {};
    for (int k0 = 0; k0 < L; k0 += 4) {
        int ak = k0 + khalf;
        v2f a;
        a.x = atn_lds[(m0 + lm) * L + ak];
        a.y = atn_lds[(m0 + lm) * L + ak + 1];
#pragma unroll
        for (int nt = 0; nt < 4; ++nt) {
            int bn = nt * 16 + lm;
            v2f b;
            b.x = v_lds[ak * 64 + bn];
            b.y = v_lds[(ak + 1) * 64 + bn];
            accO[nt] = wmma_f32_k4(a, b, accO[nt]);
        }
    }
    // scatter: merge + roll(+SH) == same pixel mapping as gather
    float* og = ocat + ((size_t)bb * CIN + chunkIdx * 64) * HWPIX;
#pragma unroll
    for (int r = 0; r < 8; ++r) {
        int m = m0 + rowhi + r;
        int p = pix(m);
#pragma unroll
        for (int nt = 0; nt < 4; ++nt)
            og[(size_t)(nt * 16 + lm) * HWPIX + p] = accO[nt][r];
    }
}

// ---------------------------------------------------------------------------
extern "C" void kernel_launch(void* const* d_in, const int* in_sizes, int n_in,
                              void* d_out, int out_size, void* d_ws, size_t ws_size,
                              hipStream_t stream) {
    const float* x     = (const float*)d_in[0];
    const float* y     = (const float*)d_in[1];
    const float* w_in  = (const float*)d_in[2];
    const float* b_in  = (const float*)d_in[3];
    const float* gamma = (const float*)d_in[4];
    const float* beta  = (const float*)d_in[5];
    const float* mean  = (const float*)d_in[6];
    const float* var   = (const float*)d_in[7];
    const float* w_out = (const float*)d_in[8];
    const float* b_out = (const float*)d_in[9];

    // workspace layout (floats): wbn | biasbn | xp | yp | ocat  (~443 MB)
    float* ws_f   = (float*)d_ws;
    float* wbn    = ws_f;
    float* biasbn = wbn + (size_t)C2 * CIN;
    float* xp     = biasbn + C2;
    float* yp     = xp + (size_t)BATCH * C2 * HWPIX;
    float* ocat   = yp + (size_t)BATCH * C2 * HWPIX;

    float* y1   = (float*)d_out;
    float* atn0 = y1 + (size_t)BATCH * CIN * HWPIX;          // 22,118,400
    float* atn1 = atn0 + (size_t)7200 * 16 * 16;             // +1,843,200
    float* atn2 = atn1 + (size_t)1800 * 64 * 64;             // +7,372,800

    // 1) fold BN into conv_in
    fold_bn_kernel<<<2, 192, 0, stream>>>(w_in, b_in, gamma, beta, mean, var,
                                          wbn, biasbn);

    // 2) conv_in + BN for x and y  (M=384, K=192)
    {
        dim3 grid(HWPIX / 64, C2 / 64, BATCH), block(128);
        gemm1x1_kernel<<<grid, block, 0, stream>>>(xp, x, wbn, biasbn, C2, CIN);
        gemm1x1_kernel<<<grid, block, 0, stream>>>(yp, y, wbn, biasbn, C2, CIN);
    }

    // 3) windowed attention, three scales
    attn_kernel<4><<<7200, 32,  (16 * 16 + 16 * 64) * sizeof(float), stream>>>(
        xp, yp, atn0, ocat, 0);
    attn_kernel<8><<<1800, 128, (64 * 64 + 64 * 64) * sizeof(float), stream>>>(
        xp, yp, atn1, ocat, 1);
    attn_kernel<12><<<800, 288, (144 * 144 + 144 * 64) * sizeof(float), stream>>>(
        xp, yp, atn2, ocat, 2);

    // 4) conv_out  (M=192, K=192)
    {
        dim3 grid(HWPIX / 64, CIN / 64, BATCH), block(128);
        gemm1x1_kernel<<<grid, block, 0, stream>>>(y1, ocat, w_out, b_out, CIN, CIN);
    }
}